// Rules_68006512165359
// MI455X (gfx1250) — compile-verified
//
#include <hip/hip_runtime.h>

// out[b,u,r] = prod_f( x[b,u,f] * rules[r,f] )
//            = (prod_f x[b,u,f]) * (prod_f rules[r,f])  -- exact factorization
//
// Phase 1: row products Px[4096], Pr[256]  -> workspace
// Phase 2: outer product Px (x) Pr via V_WMMA_F32_16X16X4_F32, one wave per
//          16x16 tile of the (4096 x 256) output.

typedef float v2f __attribute__((ext_vector_type(2)));
typedef float v8f __attribute__((ext_vector_type(8)));

#define BU_DIM 4096   // B*U = 64*64
#define R_DIM  256
#define F_DIM  64

// ---------------------------------------------------------------------------
// Kernel 1: product over the fuzzy axis for every row of x and rules.
// One thread per row; 16x float4 loads, fully unrolled multiply tree.
// ---------------------------------------------------------------------------
__global__ void rules_row_products(const float* __restrict__ x,
                                   const float* __restrict__ rules,
                                   float* __restrict__ prod) {
  const int t = blockIdx.x * blockDim.x + threadIdx.x;
  if (t >= BU_DIM + R_DIM) return;

  const float* base = (t < BU_DIM) ? (x + (size_t)t * F_DIM)
                                   : (rules + (size_t)(t - BU_DIM) * F_DIM);
  const float4* v4 = (const float4*)base;

  float p = 1.0f;
#pragma unroll
  for (int i = 0; i < F_DIM / 4; ++i) {
    float4 q = v4[i];
    p *= q.x * q.y * q.z * q.w;
  }
  prod[t] = p;
}

// ---------------------------------------------------------------------------
// Kernel 2: out[m, n] = Px[m] * Pr[n], tiled 16x16, one wave32 per tile.
//
// V_WMMA_F32_16X16X4_F32: D(16x16) = A(16x4) @ B(4x16) + C.
// We replicate Px along all K slots of A and Pr along all K slots of B
// (every lane, both VGPRs hold value[lane & 15]), which is valid for any
// half-wave-symmetric A/B layout.  Then D = 4 * Px (x) Pr; the final *0.25f
// (exact, power of two) recovers the outer product in full f32 precision.
//
// No divergence anywhere -> EXEC is all ones at the WMMA (ISA requirement).
// ---------------------------------------------------------------------------
__global__ void rules_outer_wmma(const float* __restrict__ px,
                                 const float* __restrict__ pr,
                                 float* __restrict__ out) {
  const int lane = threadIdx.x & 31;
  const int wave = threadIdx.x >> 5;
  const int tile = blockIdx.x * (blockDim.x >> 5) + wave;  // 0 .. 4095
  const int mt   = tile >> 4;   // 256 tiles along BU
  const int nt   = tile & 15;   // 16 tiles along R
  const int l16  = lane & 15;

  const float a_s = px[mt * 16 + l16];   // broadcast-friendly (halves repeat)
  const float b_s = pr[nt * 16 + l16];

  v2f a = {a_s, a_s};
  v2f b = {b_s, b_s};
  v8f c = {};
  // 8 args: (neg_a, A, neg_b, B, c_mod, C, reuse_a, reuse_b)
  v8f d = __builtin_amdgcn_wmma_f32_16x16x4_f32(
      /*neg_a=*/false, a, /*neg_b=*/false, b,
      /*c_mod=*/(short)0, c, /*reuse_a=*/false, /*reuse_b=*/false);

  // C/D layout: VGPR v holds M = v (lanes 0-15) or M = v + 8 (lanes 16-31),
  // N = lane & 15.
  const int row_base = mt * 16 + ((lane < 16) ? 0 : 8);
  const int col      = nt * 16 + l16;
#pragma unroll
  for (int v = 0; v < 8; ++v) {
    out[(size_t)(row_base + v) * R_DIM + col] = d[v] * 0.25f;
  }
}

// ---------------------------------------------------------------------------
extern "C" void kernel_launch(void* const* d_in, const int* in_sizes, int n_in,
                              void* d_out, int out_size, void* d_ws, size_t ws_size,
                              hipStream_t stream) {
  (void)in_sizes; (void)n_in; (void)out_size; (void)ws_size;

  const float* x     = (const float*)d_in[0];   // (64,64,64) f32
  const float* rules = (const float*)d_in[1];   // (256,64)   f32
  float*       out   = (float*)d_out;           // (64,64,256) f32

  float* px = (float*)d_ws;        // 4096 floats
  float* pr = px + BU_DIM;         // 256 floats   (17 KB total scratch)

  // Phase 1: 4352 row products.
  {
    const int total   = BU_DIM + R_DIM;
    const int threads = 256;
    const int blocks  = (total + threads - 1) / threads;
    rules_row_products<<<blocks, threads, 0, stream>>>(x, rules, px);
  }

  // Phase 2: 4096 16x16 tiles, 8 waves (256 threads) per block -> 512 blocks.
  {
    const int threads = 256;
    const int blocks  = (BU_DIM / 16) * (R_DIM / 16) / (threads / 32);
    rules_outer_wmma<<<blocks, threads, 0, stream>>>(px, pr, out);
  }
}